// InterAttnGateBlock_17102559773276
// MI455X (gfx1250) — compile-verified
//
#include <hip/hip_runtime.h>
#include <hip/hip_bf16.h>
#include <math.h>

typedef __attribute__((ext_vector_type(2))) float v2f;
typedef __attribute__((ext_vector_type(8))) float v8f;

#define S_LEN 1024
#define NXC   1024
#define NHEAD 16
#define HDIM  64
#define NRPE  41

__device__ __forceinline__ v8f wmma4(v2f a, v2f b, v8f c) {
  // D = A(16x4 f32) x B(4x16 f32) + C(16x16 f32)
  return __builtin_amdgcn_wmma_f32_16x16x4_f32(
      /*neg_a=*/false, a, /*neg_b=*/false, b,
      /*c_mod=*/(short)0, c, /*reuse_a=*/false, /*reuse_b=*/false);
}

// ---------------------------------------------------------------------------
// f32 WMMA GEMM, 32x32 macro-tile per wave (2x2 subtiles of 16x16).
// Compile-time leading dimensions so K-loop addresses fold into immediate
// offsets; K unrolled in chunks of 32 with a prefetch of the next B panel.
// C[M,N] = A[M,K] @ B[K,N] (+bias) (+=C if accumulate)
// ---------------------------------------------------------------------------
template <int LDA, int LDB, int LDC>
__global__ void wmma_gemm_f32(const float* __restrict__ A,
                              const float* __restrict__ Bm,
                              float* __restrict__ C,
                              int M, int N, int K,
                              const float* __restrict__ bias, int accumulate) {
  const int lane = threadIdx.x & 31;
  const int w    = threadIdx.x >> 5;
  const int tilesN = N >> 5;                       // 32-wide macro tiles
  const int tile = blockIdx.x * (blockDim.x >> 5) + w;
  if (tile >= (M >> 5) * tilesN) return;
  const int tm = tile / tilesN, tn = tile - tm * tilesN;
  const int m0 = tm << 5, n0 = tn << 5;
  const int lo = lane & 15, hi = lane >> 4;

  v8f c00 = {}, c01 = {}, c10 = {}, c11 = {};
  if (accumulate) {
#pragma unroll
    for (int r = 0; r < 8; ++r) {
      c00[r] = C[(size_t)(m0 + r + 8 * hi) * LDC + n0 + lo];
      c01[r] = C[(size_t)(m0 + r + 8 * hi) * LDC + n0 + 16 + lo];
      c10[r] = C[(size_t)(m0 + 16 + r + 8 * hi) * LDC + n0 + lo];
      c11[r] = C[(size_t)(m0 + 16 + r + 8 * hi) * LDC + n0 + 16 + lo];
    }
  }

  const float* arow0 = A + (size_t)(m0 + lo) * LDA + 2 * hi;
  const float* arow1 = arow0 + (size_t)16 * LDA;
  const float* bcol0 = Bm + (size_t)(2 * hi) * LDB + (n0 + lo);
  const float* bcol1 = bcol0 + 16;

  for (int kb = 0; kb < K; kb += 32) {
    if (kb + 32 < K) {  // prefetch next B panel chunk (global_prefetch_b8)
      __builtin_prefetch(bcol0 + (size_t)(kb + 32) * LDB, 0, 1);
      __builtin_prefetch(bcol1 + (size_t)(kb + 32) * LDB, 0, 1);
    }
#pragma unroll
    for (int u = 0; u < 8; ++u) {
      const int kk = kb + 4 * u;
      v2f a0 = *(const v2f*)(arow0 + kk);
      v2f a1 = *(const v2f*)(arow1 + kk);
      v2f b0, b1;
      b0.x = bcol0[(size_t)kk * LDB];
      b0.y = bcol0[(size_t)(kk + 1) * LDB];
      b1.x = bcol1[(size_t)kk * LDB];
      b1.y = bcol1[(size_t)(kk + 1) * LDB];
      c00 = wmma4(a0, b0, c00);
      c01 = wmma4(a0, b1, c01);
      c10 = wmma4(a1, b0, c10);
      c11 = wmma4(a1, b1, c11);
    }
  }

  float bv0 = bias ? bias[n0 + lo] : 0.0f;
  float bv1 = bias ? bias[n0 + 16 + lo] : 0.0f;
#pragma unroll
  for (int r = 0; r < 8; ++r) {
    C[(size_t)(m0 + r + 8 * hi) * LDC + n0 + lo]           = c00[r] + bv0;
    C[(size_t)(m0 + r + 8 * hi) * LDC + n0 + 16 + lo]      = c01[r] + bv1;
    C[(size_t)(m0 + 16 + r + 8 * hi) * LDC + n0 + lo]      = c10[r] + bv0;
    C[(size_t)(m0 + 16 + r + 8 * hi) * LDC + n0 + 16 + lo] = c11[r] + bv1;
  }
}

// ---------------------------------------------------------------------------
// qr[b,h,s,r] = sum_d q[b,s,h,d] * rpek[r,d]   (q lives inside qkv rows)
// ---------------------------------------------------------------------------
__global__ void qr_kernel(const float* __restrict__ qkv,
                          const float* __restrict__ rpek,
                          float* __restrict__ qr) {
  int idx = blockIdx.x * blockDim.x + threadIdx.x;
  const int total = 2 * NHEAD * S_LEN * NRPE;
  if (idx >= total) return;
  int r = idx % NRPE;
  int t = idx / NRPE;
  int s = t % S_LEN; t /= S_LEN;
  int h = t % NHEAD;
  int b = t / NHEAD;
  const float* q = qkv + ((size_t)(b * S_LEN + s)) * (3 * NXC) + h * HDIM;
  const float* wv = rpek + (size_t)r * HDIM;
  float acc = 0.f;
#pragma unroll 8
  for (int d = 0; d < HDIM; ++d) acc += q[d] * wv[d];
  qr[idx] = acc;
}

// ---------------------------------------------------------------------------
// Flash attention: one wave per (b,h,qtile16). Online softmax; scores never
// leave registers/LDS.  attn[b,s,h*64+d] output, row-major [B*S, NX].
// __launch_bounds__(128,1): keep Q/O/fragment batches in registers, no spill.
// ---------------------------------------------------------------------------
__global__ __launch_bounds__(128, 1) void flash_attn(
    const float* __restrict__ qkv,
    const float* __restrict__ qr,
    const int* __restrict__ dist,
    const float* __restrict__ mask,
    float* __restrict__ attn) {
  __shared__ float lds_p[4][256];  // one 16x16 P tile per wave
  const int lane = threadIdx.x & 31;
  const int w    = threadIdx.x >> 5;
  const int wid  = blockIdx.x * 4 + w;     // 0..2047
  const int qt = wid & 63;
  const int bh = wid >> 6;
  const int h = bh & (NHEAD - 1);
  const int b = bh >> 4;
  const int lo = lane & 15, hi = lane >> 4;
  const int q0 = qt * 16;

  // Preload Q tile in A-operand layout: 16 v2f per lane covers K=0..63
  const float* qbase = qkv + ((size_t)(b * S_LEN + q0 + lo)) * (3 * NXC) + h * HDIM + 2 * hi;
  v2f qa[16];
#pragma unroll
  for (int i = 0; i < 16; ++i) qa[i] = *(const v2f*)(qbase + 4 * i);

  v8f o0 = {}, o1 = {}, o2 = {}, o3 = {};
  float rmax[8], rsum[8];
#pragma unroll
  for (int r = 0; r < 8; ++r) { rmax[r] = -1e30f; rsum[r] = 0.f; }

  float* P = lds_p[w];

  for (int kt = 0; kt < S_LEN / 16; ++kt) {
    const int k0 = kt * 16;

    // prefetch next K/V tile rows for this lane (global_prefetch_b8)
    if (kt + 1 < S_LEN / 16) {
      const float* nk = qkv + ((size_t)(b * S_LEN + k0 + 16 + lo)) * (3 * NXC) + NXC + h * HDIM;
      __builtin_prefetch(nk, 0, 1);
      __builtin_prefetch(nk + NXC, 0, 1);  // matching V row
    }

    // ---- S = Q @ K^T : batch 8 K fragments, then 8 WMMAs, twice ----
    v8f c = {};
    const float* kbase = qkv + ((size_t)(b * S_LEN + k0 + lo)) * (3 * NXC) + NXC + h * HDIM + 2 * hi;
    {
      v2f kf[8];
#pragma unroll
      for (int i = 0; i < 8; ++i) kf[i] = *(const v2f*)(kbase + 4 * i);
#pragma unroll
      for (int i = 0; i < 8; ++i) c = wmma4(qa[i], kf[i], c);
#pragma unroll
      for (int i = 0; i < 8; ++i) kf[i] = *(const v2f*)(kbase + 32 + 4 * i);
#pragma unroll
      for (int i = 0; i < 8; ++i) c = wmma4(qa[8 + i], kf[i], c);
    }

    // ---- add RPE bias, scale, mask ----
#pragma unroll
    for (int r = 0; r < 8; ++r) {
      int qrow = q0 + r + 8 * hi;
      int kcol = k0 + lo;
      int dd = dist[((size_t)b * S_LEN + qrow) * S_LEN + kcol] + 20;
      float rpe = qr[(((size_t)(b * NHEAD + h) * S_LEN + qrow)) * NRPE + dd];
      c[r] = (c[r] + rpe) * 0.125f + mask[b * S_LEN + kcol];
    }

    // ---- online softmax (row = 16 lanes within a half-wave) ----
    float corr[8];
#pragma unroll
    for (int r = 0; r < 8; ++r) {
      float t = c[r];
      t = fmaxf(t, __shfl_xor(t, 1));
      t = fmaxf(t, __shfl_xor(t, 2));
      t = fmaxf(t, __shfl_xor(t, 4));
      t = fmaxf(t, __shfl_xor(t, 8));
      float mnew = fmaxf(rmax[r], t);
      float co = __expf(rmax[r] - mnew);
      float p = __expf(c[r] - mnew);
      c[r] = p;
      float ps = p;
      ps += __shfl_xor(ps, 1);
      ps += __shfl_xor(ps, 2);
      ps += __shfl_xor(ps, 4);
      ps += __shfl_xor(ps, 8);
      rsum[r] = rsum[r] * co + ps;
      rmax[r] = mnew;
      corr[r] = co;
    }
#pragma unroll
    for (int r = 0; r < 8; ++r) {
      o0[r] *= corr[r]; o1[r] *= corr[r]; o2[r] *= corr[r]; o3[r] *= corr[r];
    }

    // ---- transpose P through LDS: C-layout -> A-operand layout ----
#pragma unroll
    for (int r = 0; r < 8; ++r) P[(r + 8 * hi) * 16 + lo] = c[r];
    __syncthreads();
    v2f pa[4];
#pragma unroll
    for (int i = 0; i < 4; ++i) pa[i] = *(const v2f*)&P[lo * 16 + 4 * i + 2 * hi];
    __syncthreads();

    // ---- O += P @ V  (4 k-chunks; batch 4 V fragments then 4 WMMAs) ----
    const float* vbase = qkv + ((size_t)(b * S_LEN + k0)) * (3 * NXC) + 2 * NXC + h * HDIM;
#pragma unroll
    for (int i = 0; i < 4; ++i) {
      const float* vr = vbase + (size_t)(4 * i + 2 * hi) * (3 * NXC);
      v2f vb[4];
#pragma unroll
      for (int t = 0; t < 4; ++t) {
        vb[t].x = vr[t * 16 + lo];
        vb[t].y = vr[3 * NXC + t * 16 + lo];
      }
      o0 = wmma4(pa[i], vb[0], o0);
      o1 = wmma4(pa[i], vb[1], o1);
      o2 = wmma4(pa[i], vb[2], o2);
      o3 = wmma4(pa[i], vb[3], o3);
    }
  }

  // ---- finalize: divide by row sums, scatter into [B*S, NX] ----
  float* obase = attn + ((size_t)(b * S_LEN + q0)) * NXC + h * HDIM;
#pragma unroll
  for (int r = 0; r < 8; ++r) {
    float inv = 1.0f / rsum[r];
    size_t row = (size_t)(r + 8 * hi) * NXC;
    obase[row + 0 * 16 + lo] = o0[r] * inv;
    obase[row + 1 * 16 + lo] = o1[r] * inv;
    obase[row + 2 * 16 + lo] = o2[r] * inv;
    obase[row + 3 * 16 + lo] = o3[r] * inv;
  }
}

// ---------------------------------------------------------------------------
// h = LayerNorm(x + proj) * g + b   (one block of 256 per row of 1024)
// ---------------------------------------------------------------------------
__global__ void resid_ln(const float* __restrict__ x, const float* __restrict__ a,
                         const float* __restrict__ g, const float* __restrict__ bb,
                         float* __restrict__ hout) {
  __shared__ float ssum[8], ssq[8];
  const int row = blockIdx.x;
  const float* xr = x + (size_t)row * NXC;
  const float* ar = a + (size_t)row * NXC;
  float vals[4];
  float lsum = 0.f, lsq = 0.f;
#pragma unroll
  for (int i = 0; i < 4; ++i) {
    float v = xr[threadIdx.x + 256 * i] + ar[threadIdx.x + 256 * i];
    vals[i] = v; lsum += v; lsq += v * v;
  }
#pragma unroll
  for (int off = 16; off; off >>= 1) {
    lsum += __shfl_xor(lsum, off);
    lsq  += __shfl_xor(lsq, off);
  }
  const int w = threadIdx.x >> 5;
  if ((threadIdx.x & 31) == 0) { ssum[w] = lsum; ssq[w] = lsq; }
  __syncthreads();
  float tsum = 0.f, tsq = 0.f;
#pragma unroll
  for (int i = 0; i < 8; ++i) { tsum += ssum[i]; tsq += ssq[i]; }
  float mu = tsum * (1.0f / NXC);
  float var = tsq * (1.0f / NXC) - mu * mu;
  float rstd = rsqrtf(var + 1e-5f);
#pragma unroll
  for (int i = 0; i < 4; ++i) {
    int col = threadIdx.x + 256 * i;
    hout[(size_t)row * NXC + col] = (vals[i] - mu) * rstd * g[col] + bb[col];
  }
}

// ---------------------------------------------------------------------------
// out = (1-sigmoid(zpre))*h + sigmoid(zpre)*local
// ---------------------------------------------------------------------------
__global__ void gate_out(const float* __restrict__ zpre, const float* __restrict__ hbuf,
                         const float* __restrict__ local, float* __restrict__ out, int n) {
  int i = blockIdx.x * blockDim.x + threadIdx.x;
  if (i >= n) return;
  float z = 1.0f / (1.0f + __expf(-zpre[i]));
  out[i] = (1.0f - z) * hbuf[i] + z * local[i];
}

// ---------------------------------------------------------------------------
extern "C" void kernel_launch(void* const* d_in, const int* in_sizes, int n_in,
                              void* d_out, int out_size, void* d_ws, size_t ws_size,
                              hipStream_t stream) {
  const float* x     = (const float*)d_in[0];
  const float* local = (const float*)d_in[1];
  const float* mask  = (const float*)d_in[2];
  const int*   dist  = (const int*)d_in[3];
  const float* Wqkv  = (const float*)d_in[4];
  const float* bqkv  = (const float*)d_in[5];
  const float* Wproj = (const float*)d_in[6];
  const float* bproj = (const float*)d_in[7];
  const float* rpek  = (const float*)d_in[8];
  const float* ln1g  = (const float*)d_in[9];
  const float* ln1b  = (const float*)d_in[10];
  const float* gateW = (const float*)d_in[11];
  float* out = (float*)d_out;

  const size_t BS = 2 * S_LEN;                 // 2048 rows
  float* ws   = (float*)d_ws;
  float* qkv  = ws;                            // 2048*3072
  float* qr   = qkv + BS * 3 * NXC;            // 2*16*1024*41
  float* attn = qr + (size_t)2 * NHEAD * S_LEN * NRPE;
  float* proj = attn + BS * NXC;
  float* hbuf = proj + BS * NXC;
  float* zpre = hbuf + BS * NXC;

  // 1. QKV GEMM: [2048,1024]@[1024,3072]+bqkv  (64x96 macro tiles)
  wmma_gemm_f32<NXC, 3 * NXC, 3 * NXC><<<(64 * 96) / 4, 128, 0, stream>>>(
      x, Wqkv, qkv, 2048, 3 * NXC, NXC, bqkv, 0);

  // 2. qr = q . rpek^T  (tiny)
  {
    int total = 2 * NHEAD * S_LEN * NRPE;
    qr_kernel<<<(total + 255) / 256, 256, 0, stream>>>(qkv, rpek, qr);
  }

  // 3. Flash attention (2048 waves, 4 per block)
  flash_attn<<<512, 128, 0, stream>>>(qkv, qr, dist, mask, attn);

  // 4. proj GEMM: [2048,1024]@[1024,1024]+bproj  (64x32 macro tiles)
  wmma_gemm_f32<NXC, NXC, NXC><<<(64 * 32) / 4, 128, 0, stream>>>(
      attn, Wproj, proj, 2048, NXC, NXC, bproj, 0);

  // 5. h = LN(x + proj)
  resid_ln<<<2048, 256, 0, stream>>>(x, proj, ln1g, ln1b, hbuf);

  // 6. zpre = local @ gateW[:NX] + h @ gateW[NX:]
  wmma_gemm_f32<NXC, NXC, NXC><<<(64 * 32) / 4, 128, 0, stream>>>(
      local, gateW, zpre, 2048, NXC, NXC, nullptr, 0);
  wmma_gemm_f32<NXC, NXC, NXC><<<(64 * 32) / 4, 128, 0, stream>>>(
      hbuf, gateW + (size_t)NXC * NXC, zpre, 2048, NXC, NXC, nullptr, 1);

  // 7. gated output
  {
    int n = (int)(BS * NXC);
    gate_out<<<(n + 255) / 256, 256, 0, stream>>>(zpre, hbuf, local, out, n);
  }
}